// DyRepHawkes_49005576847951
// MI455X (gfx1250) — compile-verified
//
#include <hip/hip_runtime.h>
#include <math.h>

#define N      8192
#define H      128
#define BEV    100
#define KNEG   5
#define TD_MAXF 10000.0f
#define NBLK   32
#define TPB    256

typedef __attribute__((ext_vector_type(2))) float v2f;
typedef __attribute__((ext_vector_type(8))) float v8f;

// ---------------------------------------------------------------- helpers

__device__ __forceinline__ float hawkes_eval(float g, float ps, float al, float wt, float td) {
    float gp = fminf(fmaxf(g / (ps + 1e-7f), -75.0f), 75.0f);
    return ps * log1pf(expf(gp)) + al * expf(-wt * (td / TD_MAXF));
}

__device__ __forceinline__ float sigmoidf_(float x) { return 1.0f / (1.0f + expf(-x)); }

// order-preserving float<->uint mapping so shared-mem atomicMax is exact & deterministic
__device__ __forceinline__ unsigned f2ord(float f) {
    unsigned u = __float_as_uint(f);
    return (u & 0x80000000u) ? ~u : (u | 0x80000000u);
}
__device__ __forceinline__ float ord2f(unsigned u) {
    unsigned b = (u & 0x80000000u) ? (u & 0x7fffffffu) : ~u;
    return __uint_as_float(b);
}

// deterministic fixed-order tree reduction over TPB threads
__device__ __forceinline__ float blockSum(float v, float* sbuf) {
    int t = threadIdx.x;
    sbuf[t] = v; __syncthreads();
    #pragma unroll
    for (int s = TPB / 2; s > 0; s >>= 1) {
        if (t < s) sbuf[t] += sbuf[t + s];
        __syncthreads();
    }
    float r = sbuf[0]; __syncthreads();
    return r;
}

// ---------------------------------------------------------------- init: Wh_all = z0 @ Wh_w^T + Wh_b via V_WMMA_F32_16X16X4_F32
// one wave per 16x16 tile; block = 8 waves = 8 column tiles (H/16), grid = N/16 row tiles

__global__ void init_whall(const float* __restrict__ z0, const float* __restrict__ Whw,
                           const float* __restrict__ Whb, float* __restrict__ WhAll) {
    const int lane = threadIdx.x & 31;
    const int nt   = threadIdx.x >> 5;   // 0..7 : column tile
    const int mt   = blockIdx.x;         // 0..511: row tile
    const int m    = lane & 15;
    const int half = lane >> 4;

    const float* zrow = z0  + (size_t)(mt * 16 + m) * H;  // A: 16 rows of z
    const float* wrow = Whw + (size_t)(nt * 16 + m) * H;  // B = Wh_w^T tile: B[k][n] = Whw[n][k]

    v8f acc = {0.f, 0.f, 0.f, 0.f, 0.f, 0.f, 0.f, 0.f};
    #pragma unroll
    for (int kb = 0; kb < H; kb += 4) {
        const int k0 = kb + 2 * half;            // 32b A 16x4: lanes 0-15 K={0,1}, lanes 16-31 K={2,3}
        v2f a;  a.x  = zrow[k0]; a.y  = zrow[k0 + 1];
        v2f bm; bm.x = wrow[k0]; bm.y = wrow[k0 + 1];
        acc = __builtin_amdgcn_wmma_f32_16x16x4_f32(false, a, false, bm, (short)0, acc, false, false);
    }
    const int col   = nt * 16 + m;
    const float bia = Whb[col];
    #pragma unroll
    for (int c = 0; c < 8; ++c) {                // C 16x16: VGPR c -> row c (lanes 0-15) / c+8 (lanes 16-31)
        int row = mt * 16 + c + 8 * half;
        WhAll[(size_t)row * H + col] = acc[c] + bia;
    }
}

// ---------------------------------------------------------------- init: a0,a1,p0,p1 vectors

__global__ void init_vec(const float* __restrict__ z0, const float* __restrict__ om0w,
                         const float* __restrict__ om1w, float* __restrict__ avec) {
    const int i = blockIdx.x * TPB + threadIdx.x;   // 0..N-1
    const float* zr = z0 + (size_t)i * H;
    float s0 = 0.f, s1 = 0.f, s2 = 0.f, s3 = 0.f;
    #pragma unroll 4
    for (int k = 0; k < H; ++k) {
        float zz = zr[k];
        s0 += zz * om0w[k];      s1 += zz * om1w[k];
        s2 += zz * om0w[H + k];  s3 += zz * om1w[H + k];
    }
    avec[i]                 = s0;
    avec[(size_t)N + i]     = s1;
    avec[(size_t)2 * N + i] = s2;
    avec[(size_t)3 * N + i] = s3;
}

// ---------------------------------------------------------------- per-step survey: lam_sum partials + struct() partials
// partials layout (floats): [0,32) lam | [32,64) esU | [64,96) cnU | [96,128) esV | [128,160) cnV
//                           [160, 160+32*H) maxU | [160+32*H, 160+64*H) maxV

__global__ void step_scan(const float* __restrict__ z, const float* __restrict__ WhAll,
                          const float* __restrict__ avec,
                          const float* __restrict__ A, const float* __restrict__ S,
                          const float* __restrict__ t_bar, const float* __restrict__ tv,
                          const int* __restrict__ U, const int* __restrict__ V, const int* __restrict__ ET,
                          const float* __restrict__ w_t, const float* __restrict__ alpha,
                          const float* __restrict__ psi,
                          const float* __restrict__ om0b, const float* __restrict__ om1b,
                          float* __restrict__ part, int b) {
    __shared__ float    sbuf[TPB];
    __shared__ unsigned smU[H], smV[H];
    const int tid = threadIdx.x;
    const int i   = blockIdx.x * TPB + tid;        // exactly covers N
    const int u = U[b], v = V[b], et = ET[b];
    const float ps = psi[et], al = alpha[et], wt = w_t[et];
    const float bb = (et == 0) ? om0b[0] : om1b[0];
    const float* a_ = avec + (size_t)et * N;
    const float* p_ = avec + (size_t)(2 + et) * N;
    const float  tt = tv[b];
    const float* tb = t_bar + (size_t)b * N;
    const float au = a_[u], av2 = a_[v], tbu = tb[u], tbv = tb[v];

    if (tid < H) { smU[tid] = f2ord(-__builtin_inff()); smV[tid] = f2ord(-__builtin_inff()); }
    __syncthreads();

    // Hawkes survival survey (both halves; mask excludes u and v)
    float lam = 0.f;
    {
        const float tbi = tb[i], pi = p_[i];
        if (i != u && i != v) {
            lam  = hawkes_eval(au  + pi + bb, ps, al, wt, tt - fmaxf(tbu, tbi));
            lam += hawkes_eval(av2 + pi + bb, ps, al, wt, tt - fmaxf(tbv, tbi));
        }
    }

    // struct(): single-pass max of e_j * Wh_all[j,:] over neighbors (sigmoid-of-max == max-of-sigmoid)
    float eu = 0.f, ev = 0.f;
    if (A[(size_t)u * N + i] > 0.f) eu = expf(S[(size_t)u * N + i]);
    if (A[(size_t)v * N + i] > 0.f) ev = expf(S[(size_t)v * N + i]);
    if (eu > 0.f) {
        const float* wr = WhAll + (size_t)i * H;
        for (int h = 0; h < H; ++h) atomicMax(&smU[h], f2ord(eu * wr[h]));
    }
    if (ev > 0.f) {
        const float* wr = WhAll + (size_t)i * H;
        for (int h = 0; h < H; ++h) atomicMax(&smV[h], f2ord(ev * wr[h]));
    }
    __syncthreads();

    const float lamP = blockSum(lam, sbuf);
    const float esU  = blockSum(eu, sbuf);
    const float cnU  = blockSum(eu > 0.f ? 1.f : 0.f, sbuf);
    const float esV  = blockSum(ev, sbuf);
    const float cnV  = blockSum(ev > 0.f ? 1.f : 0.f, sbuf);

    const int blk = blockIdx.x;
    if (tid == 0) {
        part[blk]        = lamP;
        part[32  + blk]  = esU;
        part[64  + blk]  = cnU;
        part[96  + blk]  = esV;
        part[128 + blk]  = cnV;
    }
    float* pmU = part + 160;
    float* pmV = part + 160 + (size_t)NBLK * H;
    if (tid < H) {
        pmU[(size_t)blk * H + tid] = ord2f(smU[tid]);
        pmV[(size_t)blk * H + tid] = ord2f(smV[tid]);
    }
}

// ---------------------------------------------------------------- per-step finalize: outputs + incremental state update

__global__ void step_update(float* __restrict__ z, float* __restrict__ WhAll, float* __restrict__ avec,
                            const float* __restrict__ Whw, const float* __restrict__ Whb,
                            const float* __restrict__ Wsw, const float* __restrict__ Wsb,
                            const float* __restrict__ Wrw, const float* __restrict__ Wrb,
                            const float* __restrict__ Wtw, const float* __restrict__ Wtb,
                            const float* __restrict__ om0w, const float* __restrict__ om1w,
                            const float* __restrict__ om0b, const float* __restrict__ om1b,
                            const int* __restrict__ U, const int* __restrict__ V, const int* __restrict__ ET,
                            const int* __restrict__ NEG, const float* __restrict__ TDIFF,
                            const float* __restrict__ t_bar, const float* __restrict__ tv,
                            const float* __restrict__ w_t, const float* __restrict__ alpha,
                            const float* __restrict__ psi,
                            const float* __restrict__ part, float* __restrict__ out, int b) {
    __shared__ float hsU[H], hsV[H], zu[H], zv[H], zn0[H], zn1[H];
    __shared__ float sEsU, sCnU, sEsV, sCnV;
    const int tid = threadIdx.x;
    const int u = U[b], v = V[b], et = ET[b];
    const float* pmU = part + 160;
    const float* pmV = part + 160 + (size_t)NBLK * H;

    if (tid == 0) {
        float es = 0.f, cn = 0.f, es2 = 0.f, cn2 = 0.f;
        for (int k = 0; k < NBLK; ++k) {
            es += part[32 + k]; cn += part[64 + k]; es2 += part[96 + k]; cn2 += part[128 + k];
        }
        sEsU = es; sCnU = cn; sEsV = es2; sCnV = cn2;
    }
    __syncthreads();

    if (tid < H) {
        float mU = -__builtin_inff(), mV = -__builtin_inff();
        for (int k = 0; k < NBLK; ++k) {
            mU = fmaxf(mU, pmU[(size_t)k * H + tid]);
            mV = fmaxf(mV, pmV[(size_t)k * H + tid]);
        }
        hsU[tid] = (sCnU > 0.f) ? sigmoidf_(mU / (sEsU + 1e-7f)) : 0.f;
        hsV[tid] = (sCnV > 0.f) ? sigmoidf_(mV / (sEsV + 1e-7f)) : 0.f;
        zu[tid]  = z[(size_t)u * H + tid];
        zv[tid]  = z[(size_t)v * H + tid];
    }
    __syncthreads();

    // ---- outputs (use PRE-update a/p/z state) ----
    {
        const float ps = psi[et], al = alpha[et], wt = w_t[et];
        const float bb = (et == 0) ? om0b[0] : om1b[0];
        const float* a_ = avec + (size_t)et * N;
        const float* p_ = avec + (size_t)(2 + et) * N;
        const float* tb = t_bar + (size_t)b * N;
        const float  tt = tv[b];
        if (tid == 0) {
            float g = a_[u] + p_[v] + bb;
            out[b * 12 + 0] = hawkes_eval(g, ps, al, wt, tt - fmaxf(tb[u], tb[v]));
        } else if (tid <= 10) {
            int k = tid - 1;
            int j = NEG[b * (2 * KNEG) + k];
            float g, td;
            if (k < KNEG) { g = a_[u] + p_[j] + bb; td = tt - fmaxf(tb[u], tb[j]); }
            else          { g = a_[j] + p_[v] + bb; td = tt - fmaxf(tb[j], tb[v]); }
            out[b * 12 + 1 + k] = hawkes_eval(g, ps, al, wt, td);
        } else if (tid == 11) {
            float s = 0.f;
            for (int k = 0; k < NBLK; ++k) s += part[k];   // fixed order -> deterministic
            out[b * 12 + 11] = s;
        }
    }

    // ---- new embeddings: h0 = struct(v)Ws + z_u Wr + td[0]Wt; h1 = struct(u)Ws + z_v Wr + td[1]Wt ----
    {
        const int r = tid >> 7, d = tid & (H - 1);
        const float* hsrc = (r == 0) ? hsV : hsU;
        const float* zr   = (r == 0) ? zu  : zv;
        float acc = Wsb[d] + Wrb[d] + Wtb[d];
        #pragma unroll 4
        for (int k = 0; k < H; ++k)
            acc += hsrc[k] * Wsw[(size_t)d * H + k] + zr[k] * Wrw[(size_t)d * H + k];
        const float tsd[4] = {50.f, 7.f, 15.f, 15.f};
        #pragma unroll
        for (int j = 0; j < 4; ++j) {
            float tdn = TDIFF[((size_t)b * 2 + r) * 4 + j] / tsd[j];
            acc += tdn * Wtw[(size_t)d * 4 + j];
        }
        float znew = sigmoidf_(acc);
        if (r == 0) zn0[d] = znew; else zn1[d] = znew;
    }
    __syncthreads();

    // ---- u writes first, then v (matches .at[u].set().at[v].set() when u==v) ----
    if (tid < H) {
        z[(size_t)u * H + tid] = zn0[tid];
        float acc = Whb[tid];
        #pragma unroll 4
        for (int k = 0; k < H; ++k) acc += zn0[k] * Whw[(size_t)tid * H + k];
        WhAll[(size_t)u * H + tid] = acc;
    } else if (tid < H + 4) {
        const int vec = tid - H;  // 0:a0 1:a1 2:p0 3:p1
        const float* w = (vec == 0) ? om0w : (vec == 1) ? om1w : (vec == 2) ? (om0w + H) : (om1w + H);
        float s = 0.f;
        for (int k = 0; k < H; ++k) s += zn0[k] * w[k];
        avec[(size_t)vec * N + u] = s;
    }
    __syncthreads();
    if (tid < H) {
        z[(size_t)v * H + tid] = zn1[tid];
        float acc = Whb[tid];
        #pragma unroll 4
        for (int k = 0; k < H; ++k) acc += zn1[k] * Whw[(size_t)tid * H + k];
        WhAll[(size_t)v * H + tid] = acc;
    } else if (tid < H + 4) {
        const int vec = tid - H;
        const float* w = (vec == 0) ? om0w : (vec == 1) ? om1w : (vec == 2) ? (om0w + H) : (om1w + H);
        float s = 0.f;
        for (int k = 0; k < H; ++k) s += zn1[k] * w[k];
        avec[(size_t)vec * N + v] = s;
    }
}

// ---------------------------------------------------------------- launch

extern "C" void kernel_launch(void* const* d_in, const int* in_sizes, int n_in,
                              void* d_out, int out_size, void* d_ws, size_t ws_size,
                              hipStream_t stream) {
    const int*   U     = (const int*)d_in[0];
    const int*   V     = (const int*)d_in[1];
    const int*   ET    = (const int*)d_in[2];
    const int*   NEG   = (const int*)d_in[3];
    const float* TDIFF = (const float*)d_in[4];
    const float* t_bar = (const float*)d_in[5];
    const float* tv    = (const float*)d_in[6];
    const float* z0    = (const float*)d_in[7];
    const float* A     = (const float*)d_in[8];
    const float* S     = (const float*)d_in[9];
    const float* w_t   = (const float*)d_in[10];
    const float* alpha = (const float*)d_in[11];
    const float* psi   = (const float*)d_in[12];
    const float* om0w  = (const float*)d_in[13];
    const float* om0b  = (const float*)d_in[14];
    const float* om1w  = (const float*)d_in[15];
    const float* om1b  = (const float*)d_in[16];
    const float* Whw   = (const float*)d_in[17];
    const float* Whb   = (const float*)d_in[18];
    const float* Wsw   = (const float*)d_in[19];
    const float* Wsb   = (const float*)d_in[20];
    const float* Wrw   = (const float*)d_in[21];
    const float* Wrb   = (const float*)d_in[22];
    const float* Wtw   = (const float*)d_in[23];
    const float* Wtb   = (const float*)d_in[24];
    float* out = (float*)d_out;
    float* ws  = (float*)d_ws;

    float* z     = ws;                                   // N*H
    float* WhAll = ws + (size_t)N * H;                   // N*H
    float* avec  = ws + (size_t)2 * N * H;               // 4*N   (a0,a1,p0,p1)
    float* part  = ws + (size_t)2 * N * H + 4 * N;       // 160 + 2*32*H

    hipMemcpyAsync(z, z0, (size_t)N * H * sizeof(float), hipMemcpyDeviceToDevice, stream);
    init_whall<<<N / 16, 256, 0, stream>>>(z0, Whw, Whb, WhAll);
    init_vec<<<N / TPB, TPB, 0, stream>>>(z0, om0w, om1w, avec);

    for (int b = 0; b < BEV; ++b) {
        step_scan<<<NBLK, TPB, 0, stream>>>(z, WhAll, avec, A, S, t_bar, tv,
                                            U, V, ET, w_t, alpha, psi, om0b, om1b, part, b);
        step_update<<<1, TPB, 0, stream>>>(z, WhAll, avec, Whw, Whb, Wsw, Wsb, Wrw, Wrb,
                                           Wtw, Wtb, om0w, om1w, om0b, om1b,
                                           U, V, ET, NEG, TDIFF, t_bar, tv,
                                           w_t, alpha, psi, part, out, b);
    }
}